// FlashMultiHeadAttention_65618510348510
// MI455X (gfx1250) — compile-verified
//
#include <hip/hip_runtime.h>
#include <stdint.h>

// ---- problem constants (match reference) ----
#define B_   128
#define S_   102
#define H_   1024
#define NH_  16
#define HD_  64
#define M_   (B_ * S_)     // 13056 rows = 204 * 64
#define SP_  112           // S padded to 7*16 for WMMA tiling

typedef __attribute__((ext_vector_type(2))) float v2f;
typedef __attribute__((ext_vector_type(8))) float v8f;

static __device__ __forceinline__ v8f wmma_f32(v2f a, v2f b, v8f c) {
  // 8 args: (neg_a, A, neg_b, B, c_mod, C, reuse_a, reuse_b)
  return __builtin_amdgcn_wmma_f32_16x16x4_f32(false, a, false, b, (short)0, c,
                                               false, false);
}

// Async Global -> LDS copy, 16 bytes per lane (GLOBAL_LOAD_ASYNC_TO_LDS_B128,
// tracked by ASYNCcnt). LDS address = low 32 bits of the flat address.
static __device__ __forceinline__ void async_to_lds_b128(void* lds,
                                                         const void* gaddr) {
  uint32_t loff = (uint32_t)(uintptr_t)lds;
  asm volatile("global_load_async_to_lds_b128 %0, %1, off"
               :
               : "v"(loff), "v"(gaddr)
               : "memory");
}

template <int N>
static __device__ __forceinline__ void wait_asynccnt() {
  asm volatile("s_wait_asynccnt %0" ::"n"(N) : "memory");
}

// ---------------------------------------------------------------------------
// Pipelined WMMA-f32 GEMM core: C[tile 64x64] += X[13056,1024] @ W[1024,N]
// 256 threads (8 waves, 4x2 grid, 16x32 per wave). Double-buffered LDS tiles
// staged with async Global->LDS b128 copies.
// ---------------------------------------------------------------------------
template <int N>
static __device__ __forceinline__ void gemm_core(const float* __restrict__ X,
                                                 const float* __restrict__ W,
                                                 int m0, int n0, v8f& acc0,
                                                 v8f& acc1) {
  __shared__ float As[2][64][20];  // row stride 80B (16B aligned), pad vs banks
  __shared__ float Bs[2][16][68];  // row stride 272B (16B aligned)

  const int tid = threadIdx.x;
  const int lane = tid & 31, wave = tid >> 5;
  const int lm = lane & 15, hs = lane >> 4;
  const int wm = wave & 3, wn = wave >> 2;

  const int ar = tid >> 2, ac = (tid & 3) * 4;   // A stage: 64 x 16
  const int br = tid >> 4, bc = (tid & 15) * 4;  // B stage: 16 x 64

  const float* gA = &X[(size_t)(m0 + ar) * H_ + ac];
  const float* gB = &W[(size_t)br * N + n0 + bc];

  // Prologue: issue tile 0
  async_to_lds_b128(&As[0][ar][ac], gA);
  async_to_lds_b128(&Bs[0][br][bc], gB);

  const int NT = H_ / 16;  // 64 k-tiles
  for (int i = 0; i < NT; ++i) {
    const int cur = i & 1;
    if (i + 1 < NT) {
      // Stream next tile into the other buffer while this one is consumed.
      async_to_lds_b128(&As[cur ^ 1][ar][ac], gA + (i + 1) * 16);
      async_to_lds_b128(&Bs[cur ^ 1][br][bc], gB + (size_t)(i + 1) * 16 * N);
      wait_asynccnt<2>();  // retire tile i (async loads retire in order)
    } else {
      wait_asynccnt<0>();
    }
    __syncthreads();
#pragma unroll
    for (int k4 = 0; k4 < 16; k4 += 4) {
      const int ka = k4 + hs * 2;
      v2f a, b0, b1;
      a.x = As[cur][wm * 16 + lm][ka];
      a.y = As[cur][wm * 16 + lm][ka + 1];
      b0.x = Bs[cur][ka][wn * 32 + lm];
      b0.y = Bs[cur][ka + 1][wn * 32 + lm];
      b1.x = Bs[cur][ka][wn * 32 + 16 + lm];
      b1.y = Bs[cur][ka + 1][wn * 32 + 16 + lm];
      acc0 = wmma_f32(a, b0, acc0);
      acc1 = wmma_f32(a, b1, acc1);
    }
    __syncthreads();  // all reads of buf[cur] done before it is refilled
  }
}

// ---------------------------------------------------------------------------
// Kernel 1: QKV projection; epilogue scatters to head-major Q/K/V [B,NH,S,HD]
// ---------------------------------------------------------------------------
__global__ __launch_bounds__(256) void qkv_gemm_kernel(
    const float* __restrict__ X, const float* __restrict__ W,
    const float* __restrict__ bias, float* __restrict__ Qb,
    float* __restrict__ Kb, float* __restrict__ Vb) {
  const int tid = threadIdx.x;
  const int lane = tid & 31, wave = tid >> 5;
  const int lm = lane & 15, hs = lane >> 4;
  const int wm = wave & 3, wn = wave >> 2;
  const int m0 = blockIdx.y * 64;
  const int n0 = blockIdx.x * 64;

  v8f acc0 = {};
  v8f acc1 = {};
  gemm_core<3 * H_>(X, W, m0, n0, acc0, acc1);

  const int mb = m0 + wm * 16 + hs * 8;
  const int nc0 = n0 + wn * 32 + lm;
#pragma unroll
  for (int v = 0; v < 8; ++v) {
    const int m = mb + v;
    const int bb = m / S_;
    const int s = m - bb * S_;
#pragma unroll
    for (int t = 0; t < 2; ++t) {
      const int n = nc0 + t * 16;
      const float val = (t ? acc1[v] : acc0[v]) + bias[n];
      const int which = n >> 10;
      const int rem = n & 1023;
      const int h = rem >> 6, d = rem & 63;
      float* dst = (which == 0) ? Qb : ((which == 1) ? Kb : Vb);
      dst[(size_t)((bb * NH_ + h) * S_ + s) * HD_ + d] = val;
    }
  }
}

// ---------------------------------------------------------------------------
// Kernel 2: in-place RoPE on Q and K (head-major [B,NH,S,HD])
// ---------------------------------------------------------------------------
__global__ void rope_kernel(float* __restrict__ Qb, float* __restrict__ Kb) {
  const int total = B_ * NH_ * S_ * (HD_ / 2);
  int i = blockIdx.x * blockDim.x + threadIdx.x;
  if (i >= total) return;
  const int p = i & 31;  // pair index in head dim (HD/2 = 32)
  const int rest = i >> 5;
  const int s = rest % S_;
  const int bh = rest / S_;
  const size_t base = ((size_t)bh * S_ + s) * HD_ + 2 * p;
  // inv_freq = 10000^(-(2p)/64); logf(10000) = 9.210340371976184
  const float inv_freq =
      expf(-(float)(2 * p) * (9.210340371976184f / (float)HD_));
  const float ang = (float)s * inv_freq;
  const float c = cosf(ang), sn = sinf(ang);
  {
    const float e = Qb[base], o = Qb[base + 1];
    Qb[base] = e * c - o * sn;
    Qb[base + 1] = e * sn + o * c;
  }
  {
    const float e = Kb[base], o = Kb[base + 1];
    Kb[base] = e * c - o * sn;
    Kb[base + 1] = e * sn + o * c;
  }
}

// ---------------------------------------------------------------------------
// Kernel 3: attention for one (b,h) per workgroup.
//   scores = (Q @ K^T) * 1/sqrt(HD) -> softmax rows -> out = P @ V -> Ctx
// 256 threads = 8 waves; S padded to 112 rows; mask = all-true (reference).
// ---------------------------------------------------------------------------
__global__ __launch_bounds__(256) void attn_kernel(
    const float* __restrict__ Qb, const float* __restrict__ Kb,
    const float* __restrict__ Vb, float* __restrict__ Ctx) {
  __shared__ float Qs[SP_][68];  // row stride 272B: b128-aligned
  __shared__ float Ks[SP_][68];
  __shared__ float Vs[SP_][68];
  __shared__ float Ss[SP_][113];

  const int bh = blockIdx.x;
  const int b = bh >> 4, h = bh & 15;
  const int tid = threadIdx.x;
  const int lane = tid & 31, wave = tid >> 5;
  const int lm = lane & 15, hs = lane >> 4;
  const size_t base = (size_t)bh * S_ * HD_;

  // Stage Q, K, V via async Global->LDS (raw); zero-fill the 10 pad rows.
  for (int i = tid; i < SP_ * (HD_ / 4); i += 256) {
    const int r = i >> 4;             // row
    const int c4 = (i & 15) * 4;      // 4-float column group
    if (r < S_) {
      const size_t g = base + (size_t)r * HD_ + c4;
      async_to_lds_b128(&Qs[r][c4], Qb + g);
      async_to_lds_b128(&Ks[r][c4], Kb + g);
      async_to_lds_b128(&Vs[r][c4], Vb + g);
    } else {
      const float4 z = {0.f, 0.f, 0.f, 0.f};
      *reinterpret_cast<float4*>(&Qs[r][c4]) = z;
      *reinterpret_cast<float4*>(&Ks[r][c4]) = z;
      *reinterpret_cast<float4*>(&Vs[r][c4]) = z;
    }
  }
  wait_asynccnt<0>();
  __syncthreads();

  // Phase 1: score tiles (7x7 = 49 tiles of 16x16), K-dim = 64.
  for (int t = wave; t < 49; t += 8) {
    const int ti = t / 7, tj = t % 7;
    v8f acc = {};
#pragma unroll
    for (int k4 = 0; k4 < HD_; k4 += 4) {
      const int ka = k4 + hs * 2;
      v2f a, bf;
      a.x = Qs[ti * 16 + lm][ka];
      a.y = Qs[ti * 16 + lm][ka + 1];
      bf.x = Ks[tj * 16 + lm][ka];  // B[k][n] = K[n][k]
      bf.y = Ks[tj * 16 + lm][ka + 1];
      acc = wmma_f32(a, bf, acc);
    }
    const int n = tj * 16 + lm;
#pragma unroll
    for (int v = 0; v < 8; ++v) {
      const int m = ti * 16 + hs * 8 + v;
      Ss[m][n] = (n < S_) ? acc[v] * 0.125f : -__builtin_inff();
    }
  }
  __syncthreads();

  // Phase 2: row softmax over the 102 valid key columns.
  for (int r = tid; r < SP_; r += 256) {
    if (r < S_) {
      float mx = -__builtin_inff();
      for (int n = 0; n < S_; ++n) mx = fmaxf(mx, Ss[r][n]);
      float sum = 0.f;
      for (int n = 0; n < S_; ++n) {
        const float e = expf(Ss[r][n] - mx);
        Ss[r][n] = e;
        sum += e;
      }
      const float inv = 1.0f / sum;
      for (int n = 0; n < S_; ++n) Ss[r][n] *= inv;
      for (int n = S_; n < SP_; ++n) Ss[r][n] = 0.f;
    } else {
      for (int n = 0; n < SP_; ++n) Ss[r][n] = 0.f;
    }
  }
  __syncthreads();

  // Phase 3: out = P[112x112] @ V[112x64]  (7x4 = 28 tiles), K-dim = 112.
  for (int t = wave; t < 28; t += 8) {
    const int ti = t >> 2, tn = t & 3;
    v8f acc = {};
#pragma unroll
    for (int k4 = 0; k4 < SP_; k4 += 4) {
      const int ka = k4 + hs * 2;
      v2f a, bf;
      a.x = Ss[ti * 16 + lm][ka];
      a.y = Ss[ti * 16 + lm][ka + 1];
      bf.x = Vs[ka][tn * 16 + lm];  // B[k][n] = V[k][n]
      bf.y = Vs[ka + 1][tn * 16 + lm];
      acc = wmma_f32(a, bf, acc);
    }
    const int d = tn * 16 + lm;
#pragma unroll
    for (int v = 0; v < 8; ++v) {
      const int m = ti * 16 + hs * 8 + v;
      if (m < S_)
        Ctx[((size_t)(b * S_ + m)) * H_ + h * HD_ + d] = acc[v];
    }
  }
}

// ---------------------------------------------------------------------------
// Kernel 4: output projection  out = Ctx[13056,1024] @ W_out[1024,1024] + b
// ---------------------------------------------------------------------------
__global__ __launch_bounds__(256) void out_gemm_kernel(
    const float* __restrict__ X, const float* __restrict__ W,
    const float* __restrict__ bias, float* __restrict__ Out) {
  const int tid = threadIdx.x;
  const int lane = tid & 31, wave = tid >> 5;
  const int lm = lane & 15, hs = lane >> 4;
  const int wm = wave & 3, wn = wave >> 2;
  const int m0 = blockIdx.y * 64;
  const int n0 = blockIdx.x * 64;

  v8f acc0 = {};
  v8f acc1 = {};
  gemm_core<H_>(X, W, m0, n0, acc0, acc1);

  const int mb = m0 + wm * 16 + hs * 8;
  const int nc0 = n0 + wn * 32 + lm;
#pragma unroll
  for (int v = 0; v < 8; ++v) {
    const int m = mb + v;
#pragma unroll
    for (int t = 0; t < 2; ++t) {
      const int n = nc0 + t * 16;
      Out[(size_t)m * H_ + n] = (t ? acc1[v] : acc0[v]) + bias[n];
    }
  }
}

// ---------------------------------------------------------------------------
extern "C" void kernel_launch(void* const* d_in, const int* in_sizes, int n_in,
                              void* d_out, int out_size, void* d_ws,
                              size_t ws_size, hipStream_t stream) {
  (void)in_sizes; (void)n_in; (void)out_size; (void)ws_size;
  const float* X = (const float*)d_in[0];      // query [B,S,H]
  // d_in[1] key, d_in[2] value, d_in[3] attn_mask: unused by the reference math
  const float* W_qkv = (const float*)d_in[4];  // [H, 3H]
  const float* b_qkv = (const float*)d_in[5];  // [3H]
  const float* W_out = (const float*)d_in[6];  // [H, H]
  const float* b_out = (const float*)d_in[7];  // [H]
  float* out = (float*)d_out;

  float* ws = (float*)d_ws;
  const size_t sz = (size_t)B_ * NH_ * S_ * HD_;  // 13,369,344 floats
  float* Qb = ws;
  float* Kb = ws + sz;
  float* Vb = ws + 2 * sz;
  float* Ctx = ws + 3 * sz;

  dim3 g1(3 * H_ / 64, M_ / 64);  // 48 x 204
  qkv_gemm_kernel<<<g1, 256, 0, stream>>>(X, W_qkv, b_qkv, Qb, Kb, Vb);

  const int pairs = B_ * NH_ * S_ * (HD_ / 2);
  rope_kernel<<<(pairs + 255) / 256, 256, 0, stream>>>(Qb, Kb);

  attn_kernel<<<B_ * NH_, 256, 0, stream>>>(Qb, Kb, Vb, Ctx);

  dim3 g4(H_ / 64, M_ / 64);  // 16 x 204
  out_gemm_kernel<<<g4, 256, 0, stream>>>(Ctx, W_out, b_out, out);
}